// PositionConv2d_27032524161329
// MI455X (gfx1250) — compile-verified
//
#include <hip/hip_runtime.h>
#include <stdint.h>

// Implicit-GEMM 3x3 conv (stride 1, pad 1), fp32 WMMA (V_WMMA_F32_16X16X4_F32).
// Per filter tap kp: D[o][w] += sum_c Wgt[o][c,kp] * In[n,c,h+dh,w+dw]
//   M = out-channels (block tile 128), K = channels (32/stage, double-buffered LDS),
//   N = w (64 = full image row -> trivial padding, coalesced stores).
// 8 waves (4Mx2N), each wave 32x32 via 2x2 16x16 accumulators (max frag reuse).
// LDS staging uses gfx1250 GLOBAL_LOAD_ASYNC_TO_LDS_B32 (ASYNCcnt) overlapped with WMMA.

typedef __attribute__((ext_vector_type(2))) float v2f;
typedef __attribute__((ext_vector_type(8))) float v8f;

namespace {
constexpr int BATCH = 16;
constexpr int IN_C  = 128;
constexpr int OUT_C = 256;
constexpr int IMG_H = 64;
constexpr int IMG_W = 64;
constexpr int TAPS  = 9;
constexpr int CK    = 32;   // K (channel) chunk per stage
constexpr int ASTR  = 136;  // LDS floats per c-row, A tile (128 o + pad; 2*136%64==16)
constexpr int BSTR  = 72;   // LDS floats per c-row, B tile (64 w + pad; 2*72%64==16)
constexpr int NSTG  = TAPS * (IN_C / CK); // 36 stages
}

__device__ __forceinline__ void async_g2l(uint32_t lds_byte_off, const float* gptr) {
  // GLOBAL_LOAD_ASYNC_TO_LDS_B32: VDST = LDS byte address, VADDR = 64-bit global addr.
  asm volatile("global_load_async_to_lds_b32 %0, %1, off"
               :: "v"(lds_byte_off), "v"((uint64_t)(uintptr_t)gptr)
               : "memory");
}

__device__ __forceinline__ void wait_async_all() {
  asm volatile("s_wait_asynccnt 0x0" ::: "memory");
}

__device__ __forceinline__ void stage(const float* __restrict__ in,
                                      const float* __restrict__ wgt,
                                      int s, int n, int h, int oB, int tid,
                                      float* __restrict__ As,
                                      float* __restrict__ Bs) {
  const int kp  = s >> 2;            // filter tap
  const int cc  = (s & 3) * CK;      // channel chunk base
  const int dh  = kp / 3 - 1;
  const int dw  = kp - (kp / 3) * 3 - 1;
  const int hin = h + dh;

  // Weights -> As[c][o], 32c x 128o = 4096 elems, 16 per thread (L2-resident).
  const int a_c = tid & 31;
  const int a_o = tid >> 5;
#pragma unroll
  for (int j = 0; j < 16; ++j) {
    const int o = a_o + 8 * j;
    async_g2l((uint32_t)(uintptr_t)&As[a_c * ASTR + o],
              wgt + (size_t)(oB + o) * (IN_C * TAPS) + (size_t)(cc + a_c) * TAPS + kp);
  }

  // Input row -> Bs[c][w], 32c x 64w, 8 per thread, lane-coalesced; zero-pad OOB.
  const int b_w = tid & 63;
  const int b_c = tid >> 6;
  const int win = b_w + dw;
  const bool ok = ((unsigned)hin < (unsigned)IMG_H) && ((unsigned)win < (unsigned)IMG_W);
#pragma unroll
  for (int j = 0; j < 8; ++j) {
    const int c = b_c + 4 * j;
    float* lp = &Bs[c * BSTR + b_w];
    if (ok) {
      async_g2l((uint32_t)(uintptr_t)lp,
                in + (((size_t)n * IN_C + (cc + c)) * IMG_H + hin) * IMG_W + win);
    } else {
      *lp = 0.0f; // padding (DScnt store, covered by the barrier)
    }
  }
}

__global__ __launch_bounds__(256)
void PositionConv2d_wmma_f32(const float* __restrict__ in,
                             const float* __restrict__ wgt,
                             const float* __restrict__ bias,
                             float* __restrict__ out) {
  __shared__ float AsmB[2][CK * ASTR];
  __shared__ float BsmB[2][CK * BSTR];

  const int tid  = threadIdx.x;
  const int lane = tid & 31;   // wave32
  const int wave = tid >> 5;   // 0..7
  const int mw   = wave >> 1;  // o sub-tile (4 x 32)
  const int nw   = wave & 1;   // w half (2 x 32)
  const int half = lane >> 4;  // K-half of fragment (ISA layout)
  const int l15  = lane & 15;

  const int row = blockIdx.x;        // n*IMG_H + h
  const int n   = row >> 6;
  const int h   = row & 63;
  const int oB  = blockIdx.y * 128;  // out-channel tile base

  v8f acc00 = {}, acc01 = {}, acc10 = {}, acc11 = {};

  stage(in, wgt, 0, n, h, oB, tid, AsmB[0], BsmB[0]);
  wait_async_all();
  __syncthreads();

  for (int s = 0; s < NSTG; ++s) {
    // Kick off next stage's async DMA, then compute current while it lands.
    if (s + 1 < NSTG)
      stage(in, wgt, s + 1, n, h, oB, tid, AsmB[(s + 1) & 1], BsmB[(s + 1) & 1]);

    const float* As = AsmB[s & 1];
    const float* Bs = BsmB[s & 1];

#pragma unroll
    for (int ks = 0; ks < CK; ks += 4) {
      const int kr = ks + 2 * half; // lanes 16-31 carry K=k0+2,k0+3
      v2f a0, a1, b0, b1;
      a0[0] = As[(kr + 0) * ASTR + mw * 32 + l15];
      a0[1] = As[(kr + 1) * ASTR + mw * 32 + l15];
      a1[0] = As[(kr + 0) * ASTR + mw * 32 + 16 + l15];
      a1[1] = As[(kr + 1) * ASTR + mw * 32 + 16 + l15];
      b0[0] = Bs[(kr + 0) * BSTR + nw * 32 + l15];
      b0[1] = Bs[(kr + 1) * BSTR + nw * 32 + l15];
      b1[0] = Bs[(kr + 0) * BSTR + nw * 32 + 16 + l15];
      b1[1] = Bs[(kr + 1) * BSTR + nw * 32 + 16 + l15];
      acc00 = __builtin_amdgcn_wmma_f32_16x16x4_f32(false, a0, false, b0, (short)0, acc00, false, false);
      acc01 = __builtin_amdgcn_wmma_f32_16x16x4_f32(false, a0, false, b1, (short)0, acc01, false, false);
      acc10 = __builtin_amdgcn_wmma_f32_16x16x4_f32(false, a1, false, b0, (short)0, acc10, false, false);
      acc11 = __builtin_amdgcn_wmma_f32_16x16x4_f32(false, a1, false, b1, (short)0, acc11, false, false);
    }

    wait_async_all(); // next buffer fully landed in LDS
    __syncthreads();  // all waves done reading current buffer
  }

  // Epilogue: D layout VGPR v -> M = v + 8*half (out-channel); lane[3:0] -> N (w).
  const int wcol = nw * 32 + l15;
#pragma unroll
  for (int m2 = 0; m2 < 2; ++m2) {
    const v8f& aN0 = m2 ? acc10 : acc00;
    const v8f& aN1 = m2 ? acc11 : acc01;
#pragma unroll
    for (int v = 0; v < 8; ++v) {
      const int o = oB + mw * 32 + m2 * 16 + v + 8 * half;
      const float bv = bias[o];
      const size_t base = (((size_t)n * OUT_C + o) * IMG_H + h) * IMG_W;
      out[base + wcol]      = aN0[v] + bv;
      out[base + wcol + 16] = aN1[v] + bv;
    }
  }
}

extern "C" void kernel_launch(void* const* d_in, const int* in_sizes, int n_in,
                              void* d_out, int out_size, void* d_ws, size_t ws_size,
                              hipStream_t stream) {
  (void)in_sizes; (void)n_in; (void)out_size; (void)d_ws; (void)ws_size;
  const float* in   = (const float*)d_in[0];
  const float* wgt  = (const float*)d_in[1];
  const float* bias = (const float*)d_in[2];
  float* out        = (float*)d_out;

  dim3 grid(BATCH * IMG_H, OUT_C / 128); // 1024 x 2 blocks
  dim3 block(256);                       // 8 wave32
  PositionConv2d_wmma_f32<<<grid, block, 0, stream>>>(in, wgt, bias, out);
}